// BbRelProjection_27358941676284
// MI455X (gfx1250) — compile-verified
//
#include <hip/hip_runtime.h>
#include <hip/hip_bf16.h>

// BbRelProjection: per-row closed-form QP projections. Memory-bound streaming
// kernel: 64 B/row, ~30 FLOPs/row -> HBM-bound (~11.5us floor at 23.3 TB/s).
// MI455X strategy: aligned b128 vector memory ops, RT loads (160MB of inputs
// fit the 192MB L2 across graph replays), non-temporal b128 stores for the
// write-once output, gfx1250 global_prefetch_b8, pure 32-bit index math.

typedef float v4f __attribute__((ext_vector_type(4)));

__device__ __forceinline__ float clampf(float x, float lo, float hi) {
    return fminf(fmaxf(x, lo), hi);
}

__device__ __forceinline__ void project_row(const float* __restrict__ y,
                                            const float* __restrict__ c,
                                            float* __restrict__ o) {
    const float lx = c[0], ux = c[1], ly = c[2], uy = c[3];
    const float p0 = y[0], p1 = y[1], p2 = y[2];
    const float pa = y[3], pb = y[4], pc = y[5];

    // QP1: box clamp
    o[0] = clampf(p0, lx, ux);

    // QP2: 2-point isotonic regression then box clamp
    const float avg = 0.5f * (p1 + p2);
    const bool  swp = p1 > p2;
    o[1] = clampf(swp ? avg : p1, lx, ux);
    o[2] = clampf(swp ? avg : p2, lx, ux);

    // QP3: convex piecewise-quadratic in t, three stationary cases
    const float m     = fmaxf(pa, pb);
    const float mn    = fminf(pa, pb);
    const float t_all = (pa + pb + pc) / 3.0f;
    const float t_one = 0.5f * (pc + m);
    float t = (pc >= m) ? pc : ((t_all < mn) ? t_all : t_one);
    t = clampf(t, ly, uy);
    o[3] = clampf(pa, ly, t);
    o[4] = clampf(pb, ly, t);
    o[5] = t;
}

// nvec = nrows / 4 (threads doing the vector path); tail rows handled by tid nvec.
__global__ __launch_bounds__(256)
void bbrel_project_kernel(const float* __restrict__ yp,
                          const float* __restrict__ cp,
                          float* __restrict__ out,
                          int nvec, int nrows) {
    const int tid = blockIdx.x * 256 + threadIdx.x;

    if (tid < nvec) {
        const int r0 = tid * 4;           // rows r0..r0+3; 32-bit offsets suffice
        const int yb = r0 * 6;            // float offset into y_pred / out
        const int cb = r0 * 4;            // float offset into constr_para

        // Warm L2 ahead of demand on the cold pass (gfx1250 global_prefetch_b8).
        const int pfr = r0 + 16384;
        if (pfr + 3 < nrows) {
            __builtin_prefetch(yp + pfr * 6, 0, 1);
            __builtin_prefetch(cp + pfr * 4, 0, 1);
        }

        float y[24], c[16], o[24];

        const v4f* ypv = (const v4f*)(yp + yb);   // 96B, 16B-aligned
        #pragma unroll
        for (int i = 0; i < 6; ++i) {
            const v4f v = ypv[i];
            y[4 * i + 0] = v.x; y[4 * i + 1] = v.y;
            y[4 * i + 2] = v.z; y[4 * i + 3] = v.w;
        }

        const v4f* cpv = (const v4f*)(cp + cb);   // 64B, 16B-aligned
        #pragma unroll
        for (int i = 0; i < 4; ++i) {
            const v4f v = cpv[i];
            c[4 * i + 0] = v.x; c[4 * i + 1] = v.y;
            c[4 * i + 2] = v.z; c[4 * i + 3] = v.w;
        }

        #pragma unroll
        for (int r = 0; r < 4; ++r)
            project_row(y + 6 * r, c + 4 * r, o + 6 * r);

        // Output is written once and never re-read: non-temporal b128 stores
        // keep the 192MB L2 free for the inputs that ARE re-read across replays.
        v4f* ov = (v4f*)(out + yb);
        #pragma unroll
        for (int i = 0; i < 6; ++i) {
            v4f v;
            v.x = o[4 * i + 0]; v.y = o[4 * i + 1];
            v.z = o[4 * i + 2]; v.w = o[4 * i + 3];
            __builtin_nontemporal_store(v, ov + i);
        }
    } else if (tid == nvec) {
        // Scalar tail: at most 3 rows (not taken for BATCH = 4194304).
        for (int r = nvec * 4; r < nrows; ++r) {
            float y[6], c[4], o[6];
            #pragma unroll
            for (int i = 0; i < 6; ++i) y[i] = yp[r * 6 + i];
            #pragma unroll
            for (int i = 0; i < 4; ++i) c[i] = cp[r * 4 + i];
            project_row(y, c, o);
            #pragma unroll
            for (int i = 0; i < 6; ++i) out[r * 6 + i] = o[i];
        }
    }
}

extern "C" void kernel_launch(void* const* d_in, const int* in_sizes, int n_in,
                              void* d_out, int out_size, void* d_ws, size_t ws_size,
                              hipStream_t stream) {
    const float* yp = (const float*)d_in[0];   // y_pred:      (BATCH, 6) f32
    const float* cp = (const float*)d_in[1];   // constr_para: (BATCH, 4) f32
    float* out = (float*)d_out;                // (BATCH, 6) f32

    const int nrows = in_sizes[0] / 6;
    const int nvec  = nrows / 4;               // vector-path threads
    const int nthreads = nvec + 1;             // +1 thread for the scalar tail
    const int block = 256;                     // 8 wave32 waves per workgroup
    const int grid  = (nthreads + block - 1) / block;

    bbrel_project_kernel<<<grid, block, 0, stream>>>(yp, cp, out, nvec, nrows);
}